// ParamMemModule_3204045603526
// MI455X (gfx1250) — compile-verified
//
#include <hip/hip_runtime.h>

#define L_SZ 4096
#define HID  128
#define INNER 8
#define LRATE 0.01f
#define INV_H (2.0f / 128.0f)

// tile = 32 rows of h (16 scan steps), 16 KB
#define TILE_ROWS 32
#define TILE_BYTES (TILE_ROWS * HID * 4)
#define N_TILES (L_SZ / TILE_ROWS)   // 128

typedef float v2f  __attribute__((ext_vector_type(2)));
typedef float v8f  __attribute__((ext_vector_type(8)));
typedef unsigned int u32x4 __attribute__((ext_vector_type(4)));
typedef int   i32x4 __attribute__((ext_vector_type(4)));
typedef int   i32x8 __attribute__((ext_vector_type(8)));

// ---- VALU-pipe butterfly allreduce (no DS round trips, no s_wait_dscnt) ----
// xor1/xor2 via quad_perm DPP, xor4/xor8 via row_xmask DPP, xor16 via permlanex16.
template <int CTRL>
__device__ __forceinline__ float dpp_xor_add(float x) {
  const int xi = __float_as_int(x);
  const int yi = __builtin_amdgcn_update_dpp(xi, xi, CTRL, 0xF, 0xF, true);
  return x + __int_as_float(yi);
}

__device__ __forceinline__ float allreduce32(float x) {
  x = dpp_xor_add<0xB1>(x);          // quad_perm [1,0,3,2]  : lane ^= 1
  x = dpp_xor_add<0x4E>(x);          // quad_perm [2,3,0,1]  : lane ^= 2
  x = dpp_xor_add<0x160 | 4>(x);     // row_xmask 4          : lane ^= 4
  x = dpp_xor_add<0x160 | 8>(x);     // row_xmask 8          : lane ^= 8
  const int xi = __float_as_int(x);
  const int yi = __builtin_amdgcn_permlanex16(xi, xi, 0x76543210,
                                              (int)0xFEDCBA98, false, false); // lane ^= 16
  return x + __int_as_float(yi);     // broadcast to all 32 lanes
}

// Issue one TDM transfer: 32x128 f32 tile of h -> LDS (tracked by TENSORcnt).
// D# built per CDNA5 ISA 8.3/8.4: group0 = {flags, lds_addr, global_addr, type=2},
// group1 = {mask/data_size, tensor_dim 128x4096, tile 128x32, dim0_stride=128}.
__device__ __forceinline__ void tdm_issue_tile(unsigned long long gaddr, unsigned lds_off) {
  u32x4 g0;
  g0[0] = 1u;                                            // count=1, no gather
  g0[1] = lds_off;                                       // lds_addr (bytes)
  g0[2] = (unsigned)(gaddr & 0xFFFFFFFFull);             // global_addr[31:0]
  g0[3] = (unsigned)((gaddr >> 32) & 0x01FFFFFFull)      // global_addr[56:32]
          | (2u << 30);                                  // type = 2 ("image")
  i32x8 g1;
  g1[0] = (int)(2u << 16);        // workgroup_mask=0, data_size=2 (4 bytes)
  g1[1] = (int)(128u << 16);      // tensor_dim0 = 128 (low16 at bits 63:48)
  g1[2] = (int)(4096u << 16);     // tensor_dim0 hi16 = 0 | tensor_dim1 = 4096 (low16)
  g1[3] = (int)(128u << 16);      // tensor_dim1 hi16 = 0 | tile_dim0 = 128
  g1[4] = (int)TILE_ROWS;         // tile_dim1 = 32, tile_dim2 = 0 (2D)
  g1[5] = 128;                    // tensor_dim0_stride low32 = 128 (contiguous rows)
  g1[6] = 0;                      // dim0_stride hi16 = 0 | dim1_stride low16 = 0
  g1[7] = 8;                      // dim1_stride hi32 (524288 >> 16) -- unused for 2D
  i32x4 z4 = {0, 0, 0, 0};
#if defined(__clang_major__) && (__clang_major__ >= 23)
  i32x8 z8 = {0, 0, 0, 0, 0, 0, 0, 0};
  __builtin_amdgcn_tensor_load_to_lds(g0, g1, z4, z4, z8, 0);
#else
  __builtin_amdgcn_tensor_load_to_lds(g0, g1, z4, z4, 0);
#endif
}

__global__ __launch_bounds__(32)
void ttt_scan_kernel(const float* __restrict__ h,  const float* __restrict__ W1g,
                     const float* __restrict__ b1g, const float* __restrict__ W2g,
                     const float* __restrict__ b2g, float* __restrict__ out)
{
  __shared__ __align__(16) float tile[2 * TILE_ROWS * HID];   // 32 KB double buffer
  __shared__ __align__(16) float w2s[HID * INNER];            // epilogue W2 staging
  __shared__ __align__(16) float predbuf[HID];                // epilogue WMMA result

  const unsigned lane = threadIdx.x;   // wave32: one wave per workgroup = one batch
  const unsigned b    = blockIdx.x;

  // ---- load initial parameters into registers (lane owns HID slice 4*lane..4*lane+3)
  float w1[INNER][4];   // w1[i][j] = W1[i][4*lane+j]
  float w2[4][INNER];   // w2[j][i] = W2[4*lane+j][i]
  float b1r[INNER];     // broadcast copy
  float b2r[4];
#pragma unroll
  for (int i = 0; i < INNER; ++i) {
    const float4 t = *(const float4*)(W1g + i * HID + 4 * lane);
    w1[i][0] = t.x; w1[i][1] = t.y; w1[i][2] = t.z; w1[i][3] = t.w;
  }
#pragma unroll
  for (int j = 0; j < 4; ++j) {
    const float4 lo = *(const float4*)(W2g + (4 * lane + j) * INNER);
    const float4 hi = *(const float4*)(W2g + (4 * lane + j) * INNER + 4);
    w2[j][0] = lo.x; w2[j][1] = lo.y; w2[j][2] = lo.z; w2[j][3] = lo.w;
    w2[j][4] = hi.x; w2[j][5] = hi.y; w2[j][6] = hi.z; w2[j][7] = hi.w;
  }
#pragma unroll
  for (int i = 0; i < INNER; ++i) b1r[i] = b1g[i];
  {
    const float4 t = *(const float4*)(b2g + 4 * lane);
    b2r[0] = t.x; b2r[1] = t.y; b2r[2] = t.z; b2r[3] = t.w;
  }

  const unsigned long long hbase =
      (unsigned long long)h + (unsigned long long)b * (unsigned long long)(L_SZ * HID * 4);
  const unsigned ldsbase = (unsigned)(unsigned long long)&tile[0];

  // ---- TDM-prefetched streaming scan: double-buffered 16KB tiles
  tdm_issue_tile(hbase, ldsbase);   // tile 0 -> buffer 0
  for (int tl = 0; tl < N_TILES; ++tl) {
    if (tl < N_TILES - 1) {
      tdm_issue_tile(hbase + (unsigned long long)(tl + 1) * TILE_BYTES,
                     ldsbase + (unsigned)((tl + 1) & 1) * TILE_BYTES);
      __builtin_amdgcn_s_wait_tensorcnt(1);   // tile tl complete (in-order)
    } else {
      __builtin_amdgcn_s_wait_tensorcnt(0);
    }
    asm volatile("" ::: "memory");

    const float* buf = &tile[(tl & 1) * (TILE_ROWS * HID)];
    const int nsteps = (tl == N_TILES - 1) ? 15 : 16;   // step 2047 excluded; row 4095 = x

    for (int s = 0; s < nsteps; ++s) {
      const float4 kk = *(const float4*)(buf + (2 * s) * HID + 4 * lane);
      const float4 vv = *(const float4*)(buf + (2 * s + 1) * HID + 4 * lane);
      const float k4[4] = {kk.x, kk.y, kk.z, kk.w};
      const float v4[4] = {vv.x, vv.y, vv.z, vv.w};

      // a = W1 k + b1  (per-lane partial over 4-wide slice, VALU butterfly reduce)
      float a[INNER], r[INNER];
#pragma unroll
      for (int i = 0; i < INNER; ++i) {
        float p = w1[i][0] * k4[0];
        p = fmaf(w1[i][1], k4[1], p);
        p = fmaf(w1[i][2], k4[2], p);
        p = fmaf(w1[i][3], k4[3], p);
        a[i] = allreduce32(p) + b1r[i];
        r[i] = fmaxf(a[i], 0.0f);
      }
      // g = (2/H) * (W2 r + b2 - v)   (no reduction: r is broadcast)
      float g[4];
#pragma unroll
      for (int j = 0; j < 4; ++j) {
        float p = b2r[j];
#pragma unroll
        for (int i = 0; i < INNER; ++i) p = fmaf(w2[j][i], r[i], p);
        g[j] = INV_H * (p - v4[j]);
      }
      // da = (W2^T g) * (a > 0)
      float da[INNER];
#pragma unroll
      for (int i = 0; i < INNER; ++i) {
        float q = w2[0][i] * g[0];
        q = fmaf(w2[1][i], g[1], q);
        q = fmaf(w2[2][i], g[2], q);
        q = fmaf(w2[3][i], g[3], q);
        q = allreduce32(q);
        da[i] = (a[i] > 0.0f) ? q : 0.0f;
      }
      // SGD rank-1 updates
#pragma unroll
      for (int i = 0; i < INNER; ++i) {
        const float s1 = -LRATE * da[i];
#pragma unroll
        for (int j = 0; j < 4; ++j) w1[i][j] = fmaf(s1, k4[j], w1[i][j]);
        b1r[i] += s1;
      }
#pragma unroll
      for (int j = 0; j < 4; ++j) {
        const float s2 = -LRATE * g[j];
#pragma unroll
        for (int i = 0; i < INNER; ++i) w2[j][i] = fmaf(s2, r[i], w2[j][i]);
        b2r[j] += s2;
      }
    }
    asm volatile("" ::: "memory");
  }

  // ---- epilogue: out = W2 relu(W1 x + c1) + c2, x = h[b, L-1]
  // x = row 4095 lives in buffer (127 & 1) = 1, local row 31
  const float4 x4 = *(const float4*)(&tile[1 * TILE_ROWS * HID] + 31 * HID + 4 * lane);
  float rf[INNER];
#pragma unroll
  for (int i = 0; i < INNER; ++i) {
    float p = w1[i][0] * x4.x;
    p = fmaf(w1[i][1], x4.y, p);
    p = fmaf(w1[i][2], x4.z, p);
    p = fmaf(w1[i][3], x4.w, p);
    rf[i] = fmaxf(allreduce32(p) + b1r[i], 0.0f);
  }
  // stage final W2 to LDS as row-major [h][i]
#pragma unroll
  for (int j = 0; j < 4; ++j)
#pragma unroll
    for (int i = 0; i < INNER; ++i) w2s[(4 * lane + j) * INNER + i] = w2[j][i];
  __syncthreads();

  // pred = W2 r via V_WMMA_F32_16X16X4_F32 (full f32):
  // D = A(16x4) * B(4x16); A row 0 = r[kk], B[kk][n] = W2[16t+n][kk].
  // A layout: lanes 0-15 hold K=0,1; lanes 16-31 hold K=2,3. Only lanes 0 and 16 nonzero.
  v2f Alo = {0.0f, 0.0f}, Ahi = {0.0f, 0.0f};
  if (lane == 0)  { Alo[0] = rf[0]; Alo[1] = rf[1]; Ahi[0] = rf[4]; Ahi[1] = rf[5]; }
  if (lane == 16) { Alo[0] = rf[2]; Alo[1] = rf[3]; Ahi[0] = rf[6]; Ahi[1] = rf[7]; }
  const unsigned n     = lane & 15u;
  const unsigned kkoff = (lane < 16) ? 0u : 2u;
#pragma unroll
  for (int t = 0; t < 8; ++t) {
    const float* wr = &w2s[(16 * t + n) * INNER + kkoff];
    v2f Blo = {wr[0], wr[1]};
    v2f Bhi = {wr[4], wr[5]};
    v8f c = {0.f, 0.f, 0.f, 0.f, 0.f, 0.f, 0.f, 0.f};
    c = __builtin_amdgcn_wmma_f32_16x16x4_f32(false, Alo, false, Blo, (short)0, c, false, false);
    c = __builtin_amdgcn_wmma_f32_16x16x4_f32(false, Ahi, false, Bhi, (short)0, c, false, false);
    // row 0 of D: VGPR 0, lanes 0..15 hold N=0..15 -> pred[16t+n]
    if (lane < 16) predbuf[16 * t + lane] = c[0];
  }
  __syncthreads();

  const float4 p4 = *(const float4*)(&predbuf[4 * lane]);
  float4 o;
  o.x = p4.x + b2r[0]; o.y = p4.y + b2r[1]; o.z = p4.z + b2r[2]; o.w = p4.w + b2r[3];
  *(float4*)(out + b * HID + 4 * lane) = o;
}

extern "C" void kernel_launch(void* const* d_in, const int* in_sizes, int n_in,
                              void* d_out, int out_size, void* d_ws, size_t ws_size,
                              hipStream_t stream) {
  (void)n_in; (void)out_size; (void)d_ws; (void)ws_size;
  const float* h  = (const float*)d_in[0];
  const float* W1 = (const float*)d_in[1];
  const float* b1 = (const float*)d_in[2];
  const float* W2 = (const float*)d_in[3];
  const float* b2 = (const float*)d_in[4];
  float* out = (float*)d_out;
  const int Bn = in_sizes[0] / (L_SZ * HID);   // 256
  ttt_scan_kernel<<<dim3((unsigned)Bn), dim3(32), 0, stream>>>(h, W1, b1, W2, b2, out);
}